// ChamferLoss_11665131176430
// MI455X (gfx1250) — compile-verified
//
#include <hip/hip_runtime.h>
#include <math.h>

typedef __attribute__((ext_vector_type(2))) float v2f;
typedef __attribute__((ext_vector_type(8))) float v8f;

#define B_ 16
#define N_ 4096
#define NSTRIP (B_ * (N_ / 16))            /* 4096 strips  */
#define INVBN (1.0f / (16.0f * 4096.0f))
#define LDS_BYTES (N_ * 16 + N_ * 4) /* 64KB padded preds + 16KB colmin */

#define GLOBAL_AS __attribute__((address_space(1)))
#define LDS_AS __attribute__((address_space(3)))

#if __has_builtin(__builtin_amdgcn_global_load_async_to_lds_b32)
#define HAVE_ASYNC_LDS 1
#else
#define HAVE_ASYNC_LDS 0
#endif

__global__ void chamfer_init(float* __restrict__ colmin) {
  int i = blockIdx.x * blockDim.x + threadIdx.x;
  if (i < B_ * N_) colmin[i] = INFINITY;
}

// Block = 8 wave32s, all strips of the same batch. Stage the batch's preds
// (4096 pts, {x,y,z,yy} 16B slots) into LDS via async global->LDS DMA, then
// each wave sweeps 256 column tiles. The distance xx+yy-2*g.p is produced
// DIRECTLY by V_WMMA_F32_16X16X4_F32: A=(-2gx,-2gy,-2gz,1), B=(px,py,pz,yy),
// C=broadcast(xx). Column mins combine through LDS ds_min_num_f32.
__global__ __launch_bounds__(256) void chamfer_main(
    const float* __restrict__ preds, const float* __restrict__ gts,
    float* __restrict__ colmin, float* __restrict__ rowsums) {
  extern __shared__ char smem[];
  float* s_preds = (float*)smem;                   // [N_][4]: x,y,z,yy
  float* s_colmin = (float*)(smem + N_ * 16);      // [N_]

  const int wave = threadIdx.x >> 5;
  const int lane = threadIdx.x & 31;
  const int strip = blockIdx.x * 8 + wave;  // 4096 strips total
  const int b = strip >> 8;                 // batch (uniform within block)
  const int i0 = (strip & 255) << 4;        // row-tile base
  const int half = lane >> 4;               // K={0,1} vs K={2,3} lane group
  const int n = lane & 15;

  // ---- Stage preds[b] into LDS (AoS 12B -> 16B slots, yy filled later) ----
  const char* src = (const char*)(preds + (size_t)b * N_ * 3);
#pragma unroll
  for (int k = 0; k < 16; ++k) {
    const int j = threadIdx.x + (k << 8);
    s_colmin[j] = INFINITY;
#if HAVE_ASYNC_LDS
    __builtin_amdgcn_global_load_async_to_lds_b32(
        (GLOBAL_AS int*)(src + (size_t)j * 12 + 0),
        (LDS_AS int*)(s_preds + j * 4 + 0), 0, 0);
    __builtin_amdgcn_global_load_async_to_lds_b32(
        (GLOBAL_AS int*)(src + (size_t)j * 12 + 4),
        (LDS_AS int*)(s_preds + j * 4 + 1), 0, 0);
    __builtin_amdgcn_global_load_async_to_lds_b32(
        (GLOBAL_AS int*)(src + (size_t)j * 12 + 8),
        (LDS_AS int*)(s_preds + j * 4 + 2), 0, 0);
#else
    const float* p = (const float*)(src + (size_t)j * 12);
    s_preds[j * 4 + 0] = p[0];
    s_preds[j * 4 + 1] = p[1];
    s_preds[j * 4 + 2] = p[2];
#endif
  }
#if HAVE_ASYNC_LDS
#if __has_builtin(__builtin_amdgcn_s_wait_asynccnt)
  __builtin_amdgcn_s_wait_asynccnt(0);
#else
  asm volatile("s_wait_asynccnt 0x0" ::: "memory");
#endif
#endif
  // Fill yy = |p|^2 into each slot's 4th word (own async data, post-wait).
#pragma unroll
  for (int k = 0; k < 16; ++k) {
    const int j = threadIdx.x + (k << 8);
    float px = s_preds[j * 4 + 0];
    float py = s_preds[j * 4 + 1];
    float pz = s_preds[j * 4 + 2];
    s_preds[j * 4 + 3] = fmaf(px, px, fmaf(py, py, pz * pz));
  }
  __syncthreads();

  // ---- A fragment: rows i0..i0+15 as (-2gx,-2gy,-2gz,1) ----
  const float* g = gts + ((size_t)b * N_ + i0 + n) * 3;
  float gx = g[0], gy = g[1], gz = g[2];
  float xx = fmaf(gx, gx, fmaf(gy, gy, gz * gz));
  v2f a;
  a.x = -2.0f * (half ? gz : gx);     // K = 2 : 0
  a.y = half ? 1.0f : -2.0f * gy;     // K = 3(ones) : 1

  // C operand: broadcast xx so c[v] = xx of row (v + half*8); reused per tile.
  v8f xxc;
#pragma unroll
  for (int v = 0; v < 8; ++v) xxc[v] = __shfl(xx, v + (half << 3), 32);

  float rmin[8];
#pragma unroll
  for (int v = 0; v < 8; ++v) rmin[v] = INFINITY;

  // ---- Sweep all 256 column tiles: WMMA emits distances directly ----
  const char* spbase = (const char*)s_preds + (half << 3);
  for (int j0 = 0; j0 < N_; j0 += 16) {
    // half0 lanes read (px,py) = K0,K1; half1 lanes read (pz,yy) = K2,K3.
    v2f bb = *(const v2f*)(spbase + (size_t)(j0 + n) * 16);

    v8f c = __builtin_amdgcn_wmma_f32_16x16x4_f32(false, a, false, bb,
                                                  (short)0, xxc, false, false);

    float cmin = INFINITY;
#pragma unroll
    for (int v = 0; v < 8; ++v) {
      rmin[v] = fminf(rmin[v], c[v]);
      cmin = fminf(cmin, c[v]);
    }
    // All 32 lanes combine into the block-level LDS column-min array.
    __hip_atomic_fetch_min(&s_colmin[j0 + n], cmin, __ATOMIC_RELAXED,
                           __HIP_MEMORY_SCOPE_WORKGROUP);
  }

  // ---- Exact row mins: reduce across the 16 lanes of each half ----
  float rsum = 0.0f;
#pragma unroll
  for (int v = 0; v < 8; ++v) {
#pragma unroll
    for (int o = 1; o < 16; o <<= 1)
      rmin[v] = fminf(rmin[v], __shfl_xor(rmin[v], o, 32));
    rsum += rmin[v];
  }
  // Deterministic: store per-strip partial sums (lane 0: rows 0-7, lane 16:
  // rows 8-15); finalize adds them in fixed order.
  if (n == 0) rowsums[strip * 2 + half] = rsum;

  // ---- Flush block-level column mins to global (one atomic per column) ----
  __syncthreads();
#pragma unroll
  for (int k = 0; k < 16; ++k) {
    const int cidx = threadIdx.x + (k << 8);
    __hip_atomic_fetch_min(&colmin[(size_t)b * N_ + cidx], s_colmin[cidx],
                           __ATOMIC_RELAXED, __HIP_MEMORY_SCOPE_AGENT);
  }
}

__global__ __launch_bounds__(1024) void chamfer_final(
    const float* __restrict__ colmin, const float* __restrict__ rowsums,
    float* __restrict__ out) {
  __shared__ float sm[1024];
  float s = 0.0f;
  for (int i = threadIdx.x; i < B_ * N_; i += 1024) s += colmin[i];
  for (int i = threadIdx.x; i < NSTRIP * 2; i += 1024) s += rowsums[i];
  sm[threadIdx.x] = s;
  __syncthreads();
  for (int off = 512; off > 0; off >>= 1) {
    if (threadIdx.x < off) sm[threadIdx.x] += sm[threadIdx.x + off];
    __syncthreads();
  }
  if (threadIdx.x == 0) out[0] = sm[0] * INVBN;
}

extern "C" void kernel_launch(void* const* d_in, const int* in_sizes, int n_in,
                              void* d_out, int out_size, void* d_ws,
                              size_t ws_size, hipStream_t stream) {
  const float* preds = (const float*)d_in[0];
  const float* gts = (const float*)d_in[1];

  float* colmin = (float*)d_ws;                             // B_*N_ floats
  float* rowsums = (float*)d_ws + (size_t)B_ * N_;          // 2*NSTRIP floats

  chamfer_init<<<(B_ * N_ + 255) / 256, 256, 0, stream>>>(colmin);
  chamfer_main<<<NSTRIP / 8, 256, LDS_BYTES, stream>>>(preds, gts, colmin,
                                                       rowsums);
  chamfer_final<<<1, 1024, 0, stream>>>(colmin, rowsums, (float*)d_out);
}